// LeapfrogIntegrator_40080634806920
// MI455X (gfx1250) — compile-verified
//
#include <hip/hip_runtime.h>
#include <hip/hip_bf16.h>

// Fused leapfrog integrator for MI455X (gfx1250), wave32 + V_WMMA_F32_16X16X4_F32.
// Round 2: k-outer/ntile-inner GEMM loops (A-fragment reuse -> lower LDS operand
// bandwidth per WMMA), boundary features precomputed once per half-step into an
// LDS buffer shared with the S staging tile, force tile dropped from LDS
// (re-read from L2-resident global).

#define DD 64
#define RR 32

typedef __attribute__((ext_vector_type(2))) float v2f;
typedef __attribute__((ext_vector_type(4))) float v4f;
typedef __attribute__((ext_vector_type(8))) float v8f;

__device__ __forceinline__ v8f wmma4(v2f a, v2f b, v8f c) {
  // D = A(16x4,f32) * B(4x16,f32) + C(16x16,f32)
  return __builtin_amdgcn_wmma_f32_16x16x4_f32(
      /*neg_a=*/false, a, /*neg_b=*/false, b,
      /*c_mod=*/(short)0, c, /*reuse_a=*/false, /*reuse_b=*/false);
}

__global__ __launch_bounds__(128) void leapfrog_wmma_kernel(
    const float* __restrict__ gX, const float* __restrict__ gV,
    const float* __restrict__ gF, const float* __restrict__ gU,
    const float* __restrict__ gW, const float* __restrict__ gWf,
    const float* __restrict__ gbf, const float* __restrict__ gWi,
    const int* __restrict__ gSteps, float* __restrict__ gOut, int Btot) {

  // ---- block-shared weights (B-operands: weight[n][k] row-major == B^T) ----
  __shared__ __align__(16) float sW [RR * DD];        // W [32][64]
  __shared__ __align__(16) float sU [DD * RR];        // U [64][32]
  __shared__ __align__(16) float sWf[DD * 2 * DD];    // Wf[64][128]
  __shared__ __align__(16) float sWi[DD * DD];        // Wi[64][64]
  __shared__ __align__(16) float sbf[DD];
  // ---- per-wave state tiles ----
  __shared__ __align__(16) float sX  [4][16 * DD];     // positions [16][64]
  __shared__ __align__(16) float sV  [4][16 * DD];     // velocities [16][64]
  __shared__ __align__(16) float sFI [4][16 * DD];     // force @ Wi^T (loop-invariant)
  __shared__ __align__(16) float sFt [4][16 * 2 * DD]; // feat [16][128]; first 2KB doubles as S [16][32]
  __shared__ float sNrm[4][16];

  const int tid  = threadIdx.x;
  const int wv   = tid >> 5;
  const int lane = tid & 31;
  const int hh   = lane >> 4;   // lane half (K split)
  const int l16  = lane & 15;

  // cooperative weight load
  for (int i = tid; i < RR * DD; i += 128) sW[i]  = gW[i];
  for (int i = tid; i < DD * RR; i += 128) sU[i]  = gU[i];
  for (int i = tid; i < DD * 2 * DD; i += 128) sWf[i] = gWf[i];
  for (int i = tid; i < DD * DD; i += 128) sWi[i] = gWi[i];
  if (tid < DD) sbf[tid] = gbf[tid];

  // per-wave tile: 16 rows of x, v
  const long rowBase = (long)blockIdx.x * 64 + (long)wv * 16;
  const long gbase   = rowBase * DD;
  for (int i = lane; i < 16 * DD / 4; i += 32) {
    ((v4f*)sX[wv])[i] = ((const v4f*)(gX + gbase))[i];
    ((v4f*)sV[wv])[i] = ((const v4f*)(gV + gbase))[i];
  }
  __syncthreads();

  // ---- FI = force @ Wi^T (once; force never changes). A-frags from global. ----
  {
    const float* fTile = gF + gbase;
    v8f c[4] = {{}, {}, {}, {}};
    for (int k0 = 0; k0 < DD; k0 += 4) {
      v2f a = *(const v2f*)&fTile[l16 * DD + k0 + 2 * hh];
#pragma unroll
      for (int nt = 0; nt < 4; ++nt) {
        v2f b = *(const v2f*)&sWi[(nt * 16 + l16) * DD + k0 + 2 * hh];
        c[nt] = wmma4(a, b, c[nt]);
      }
    }
#pragma unroll
    for (int nt = 0; nt < 4; ++nt)
#pragma unroll
      for (int r = 0; r < 8; ++r)
        sFI[wv][(r + 8 * hh) * DD + nt * 16 + l16] = c[nt][r];
  }

  const int   nsteps  = gSteps[0];
  const float eff_dt  = 0.01f;           // DT * DT_SCALE
  const float hdt     = 0.5f * eff_dt;
  const float PIf     = 3.14159265358979323846f;
  const float TWO_PIf = 6.28318530717958647692f;

  for (int hs = 0; hs < 2 * nsteps; ++hs) {
    const bool update_x = (hs & 1) == 0;

    // ---- pv = v@W^T, px = x@W^T ; S = pv^2 * tanh(px) (S -> sFt[0:512)) ----
    {
      v8f cpv[2] = {{}, {}};
      v8f cpx[2] = {{}, {}};
      for (int k0 = 0; k0 < DD; k0 += 4) {
        v2f av = *(const v2f*)&sV[wv][l16 * DD + k0 + 2 * hh];
        v2f ax = *(const v2f*)&sX[wv][l16 * DD + k0 + 2 * hh];
#pragma unroll
        for (int nt = 0; nt < 2; ++nt) {
          v2f b = *(const v2f*)&sW[(nt * 16 + l16) * DD + k0 + 2 * hh];
          cpv[nt] = wmma4(av, b, cpv[nt]);
          cpx[nt] = wmma4(ax, b, cpx[nt]);
        }
      }
#pragma unroll
      for (int nt = 0; nt < 2; ++nt)
#pragma unroll
        for (int r = 0; r < 8; ++r) {
          float s = cpv[nt][r] * cpv[nt][r] * tanhf(cpx[nt][r]);
          sFt[wv][(r + 8 * hh) * RR + nt * 16 + l16] = s;   // S staging
        }
    }

    // ---- gamma = S @ U^T ----
    v8f gacc[4] = {{}, {}, {}, {}};
    for (int k0 = 0; k0 < RR; k0 += 4) {
      v2f a = *(const v2f*)&sFt[wv][l16 * RR + k0 + 2 * hh];  // S A-frag
#pragma unroll
      for (int nt = 0; nt < 4; ++nt) {
        v2f b = *(const v2f*)&sU[(nt * 16 + l16) * RR + k0 + 2 * hh];
        gacc[nt] = wmma4(a, b, gacc[nt]);
      }
    }

    // ---- per-row velocity norm of current sV ----
    {
      float ss = 0.f;
      const float* vr = &sV[wv][l16 * DD + 32 * hh];
      for (int d = 0; d < 32; ++d) { float t = vr[d]; ss += t * t; }
      ss += __shfl_xor(ss, 16);
      float vn = sqrtf(ss) * (1.0f / (8.0f + 1e-6f)); // sqrt(D)=8, EPS_SMOOTH
      if (lane < 16) sNrm[wv][l16] = vn;
    }

    // ---- feat = [sin(x), cos(x)] once per half-step (S is dead now) ----
    {
      const float* xs = &sX[wv][lane * 32];                 // 32 elems per lane
      float* ftS = &sFt[wv][(lane >> 1) * (2 * DD) + (lane & 1) * 32];
      float* ftC = ftS + DD;
#pragma unroll
      for (int j = 0; j < 32; j += 4) {
        v4f xx = *(const v4f*)&xs[j];
        v4f sv, cv;
        sv.x = __sinf(xx.x); sv.y = __sinf(xx.y); sv.z = __sinf(xx.z); sv.w = __sinf(xx.w);
        cv.x = __cosf(xx.x); cv.y = __cosf(xx.y); cv.z = __cosf(xx.z); cv.w = __cosf(xx.w);
        *(v4f*)&ftS[j] = sv;
        *(v4f*)&ftC[j] = cv;
      }
    }

    // ---- gate = feat @ Wf^T ----
    v8f macc[4] = {{}, {}, {}, {}};
    for (int k0 = 0; k0 < 2 * DD; k0 += 4) {
      v2f a = *(const v2f*)&sFt[wv][l16 * (2 * DD) + k0 + 2 * hh];
#pragma unroll
      for (int nt = 0; nt < 4; ++nt) {
        v2f b = *(const v2f*)&sWf[(nt * 16 + l16) * (2 * DD) + k0 + 2 * hh];
        macc[nt] = wmma4(a, b, macc[nt]);
      }
    }

    // ---- elementwise half-kick (+ drift & torus wrap on even half-steps) ----
    {
      const float* fTile = gF + gbase;   // force stays in L2
#pragma unroll
      for (int nt = 0; nt < 4; ++nt)
#pragma unroll
        for (int r = 0; r < 8; ++r) {
          const int row = r + 8 * hh;
          const int col = nt * 16 + l16;
          float gate = macc[nt][r] + sbf[col] + sFI[wv][row * DD + col];
          float sig  = 1.0f / (1.0f + __expf(-gate));
          float mu   = sig * 0.02f * (1.0f + 0.1f * sNrm[wv][row]);
          float fo   = fTile[row * DD + col];
          float vcur = sV[wv][row * DD + col];
          float vh   = (vcur + hdt * (fo - gacc[nt][r])) / (1.0f + hdt * mu + 1e-7f);
          sV[wv][row * DD + col] = vh;
          if (update_x) {
            float xn = sX[wv][row * DD + col] + eff_dt * vh + PIf;
            xn = xn - TWO_PIf * floorf(xn * (1.0f / TWO_PIf)); // python-style mod
            sX[wv][row * DD + col] = xn - PIf;
          }
        }
    }
  }

  // ---- store results: out = [x_flat | v_flat] ----
  float* outX = gOut;
  float* outV = gOut + (long)Btot * DD;
  for (int i = lane; i < 16 * DD / 4; i += 32) {
    ((v4f*)(outX + gbase))[i] = ((v4f*)sX[wv])[i];
    ((v4f*)(outV + gbase))[i] = ((v4f*)sV[wv])[i];
  }
}

extern "C" void kernel_launch(void* const* d_in, const int* in_sizes, int n_in,
                              void* d_out, int out_size, void* d_ws, size_t ws_size,
                              hipStream_t stream) {
  const float* x  = (const float*)d_in[0];
  const float* v  = (const float*)d_in[1];
  const float* f  = (const float*)d_in[2];
  const float* U  = (const float*)d_in[3];
  const float* W  = (const float*)d_in[4];
  const float* Wf = (const float*)d_in[5];
  const float* bf = (const float*)d_in[6];
  const float* Wi = (const float*)d_in[7];
  const int* steps = (const int*)d_in[8];
  float* out = (float*)d_out;

  const int Btot = in_sizes[0] / DD;   // 131072
  const int grid = Btot / 64;          // 64 rows per workgroup (4 waves x 16)

  leapfrog_wmma_kernel<<<grid, 128, 0, stream>>>(x, v, f, U, W, Wf, bf, Wi,
                                                 steps, out, Btot);
}